// CombinedLoss_90804198572789
// MI455X (gfx1250) — compile-verified
//
#include <hip/hip_runtime.h>
#include <math.h>
#include <stdint.h>

// Problem shape (from reference setup_inputs): logits (8,1,512,512) f32, target i32.
#define B_  8
#define H_  512
#define W_  512
#define NB_ (H_ * W_)      // 262144 elements per batch image
#define N_  (B_ * NB_)     // 2097152 total elements
#define BIGF 1024.0f       // H + W, matches reference BIG

// Reduction partial counts
#define EW_BLOCKS_PER_B 64
#define EW_BLOCKS (B_ * EW_BLOCKS_PER_B)        // 512 blocks, 4096 elems each
#define ROWS_ (B_ * H_)                          // 4096 rows for boundary pass

// ---------------- CDNA5 feature availability (probe via __has_builtin) -------
#if defined(__AMDGCN__) && defined(__has_builtin)
#  if __has_builtin(__builtin_amdgcn_tensor_load_to_lds) && __has_builtin(__builtin_amdgcn_s_wait_tensorcnt)
#    define HAVE_TDM 1
#  else
#    define HAVE_TDM 0
#  endif
#  if __has_builtin(__builtin_amdgcn_wmma_f32_16x16x4_f32)
#    define HAVE_WMMA 1
#  else
#    define HAVE_WMMA 0
#  endif
#else
#  define HAVE_TDM 0
#  define HAVE_WMMA 0
#endif

#if defined(__HIP_DEVICE_COMPILE__)
#  if !HAVE_TDM
#    warning "CDNA5 probe: tensor_load_to_lds builtin UNAVAILABLE -> fallback LDS loads"
#  endif
#  if !HAVE_WMMA
#    warning "CDNA5 probe: wmma_f32_16x16x4_f32 builtin UNAVAILABLE -> scalar finalize"
#  endif
#endif

typedef unsigned int u32x4 __attribute__((ext_vector_type(4)));
typedef int          i32x4 __attribute__((ext_vector_type(4)));
typedef int          i32x8 __attribute__((ext_vector_type(8)));
typedef float        v2f   __attribute__((ext_vector_type(2)));
typedef float        v8f   __attribute__((ext_vector_type(8)));

// =============================================================================
// Kernel 1: column pass of the exact EDT.
// g2[idx] <- min(d_up, d_dn, BIG)^2 per reference (all values exact in f32).
// One thread per (b, w) column; fully coalesced (consecutive lanes = consecutive w).
// =============================================================================
__global__ void edt_cols_kernel(const int* __restrict__ tgt, float* __restrict__ g2) {
  int t = blockIdx.x * blockDim.x + threadIdx.x;   // 0 .. B_*W_-1
  int b = t >> 9;                                   // t / 512
  int w = t & (W_ - 1);
  const int base = b * NB_ + w;

  // Downward scan: distance to nearest on-pixel above (or h + 1024 if none).
  float last = -BIGF;
  for (int h = 0; h < H_; ++h) {
    int idx = base + h * W_;
    if (tgt[idx] > 0) last = (float)h;
    g2[idx] = (float)h - last;
  }
  // Upward scan: combine with distance below, cap at BIG, square.
  float nxt = 2.0f * BIGF;
  for (int h = H_ - 1; h >= 0; --h) {
    int idx = base + h * W_;
    if (tgt[idx] > 0) nxt = (float)h;
    float g = fminf(fminf(g2[idx], nxt - (float)h), BIGF);
    g2[idx] = g * g;
  }
}

// =============================================================================
// Kernel 2: elementwise pass -> per-block partials for ce / inter / sum_p / sum_t.
// Each block covers a contiguous 4096-element chunk inside exactly one batch.
// Fixed-order tree reductions => deterministic.
// =============================================================================
__global__ void ew_reduce_kernel(const float* __restrict__ logits,
                                 const int*   __restrict__ tgt,
                                 float* __restrict__ ce_part,
                                 float* __restrict__ inter_part,
                                 float* __restrict__ sump_part,
                                 float* __restrict__ sumt_part) {
  __shared__ float s0[256], s1[256], s2[256], s3[256];
  const int blk  = blockIdx.x;            // 0 .. EW_BLOCKS-1
  const int base = blk * 4096;
  float ce = 0.f, inter = 0.f, sp = 0.f, st = 0.f;
  for (int i = threadIdx.x; i < 4096; i += 256) {
    const int idx = base + i;
    const float x = logits[idx];
    float tv = (float)tgt[idx];
    tv = fminf(fmaxf(tv, 0.f), 1.f);
    // stable BCE-with-logits: max(x,0) - x*t + log1p(exp(-|x|))
    ce += fmaxf(x, 0.f) - x * tv + log1pf(expf(-fabsf(x)));
    float p = 1.0f / (1.0f + expf(-x));
    p = fminf(fmaxf(p, 1e-6f), 1.0f - 1e-6f);
    inter += p * tv;
    sp    += p;
    st    += tv;
  }
  const int tid = threadIdx.x;
  s0[tid] = ce; s1[tid] = inter; s2[tid] = sp; s3[tid] = st;
  __syncthreads();
  for (int s = 128; s > 0; s >>= 1) {
    if (tid < s) {
      s0[tid] += s0[tid + s];
      s1[tid] += s1[tid + s];
      s2[tid] += s2[tid + s];
      s3[tid] += s3[tid + s];
    }
    __syncthreads();
  }
  if (tid == 0) {
    ce_part[blk]    = s0[0];
    inter_part[blk] = s1[0];
    sump_part[blk]  = s2[0];
    sumt_part[blk]  = s3[0];
  }
}

// =============================================================================
// Kernel 3: per-batch empty-mask flags (reference: dt = 0 when mask empty).
// sums of exact 0/1 floats -> exact; == 0 compare is safe.
// =============================================================================
__global__ void flags_kernel(const float* __restrict__ sumt_part, float* __restrict__ flags) {
  int b = threadIdx.x;
  if (b < B_) {
    float s = 0.f;
    for (int i = 0; i < EW_BLOCKS_PER_B; ++i) s += sumt_part[b * EW_BLOCKS_PER_B + i];
    flags[b] = (s == 0.f) ? 0.f : 1.f;
  }
}

// =============================================================================
// Kernel 4: row lower-envelope + boundary accumulation.
// One block (256 thr, 8 wave32) per image row. The 2KB g2 row tile is DMA'd
// into LDS with the gfx1250 Tensor Data Mover, then each thread computes
// dist2[x] = min_j (x-j)^2 + g2[j] with an outward early-exit scan (exact
// same min as the reference brute force; all terms are small exact integers).
// =============================================================================
__global__ void boundary_kernel(const float* __restrict__ logits,
                                const float* __restrict__ g2,
                                const float* __restrict__ flags,
                                float* __restrict__ bpart) {
  __shared__ float g2row[W_];
  __shared__ float red[256];
  const int row = blockIdx.x;           // 0 .. ROWS_-1 == b*H + h
  const int b   = row >> 9;             // row / H_

#if HAVE_TDM
  {
    // Build Tensor DMA Descriptor (D#): 1-row tile, 512 x 4B elements.
    const unsigned int ldsoff =
        (unsigned int)(unsigned long long)(uintptr_t)(&g2row[0]); // LDS byte offset
    const unsigned long long ga =
        (unsigned long long)(uintptr_t)(const void*)(g2 + (size_t)row * W_);

    u32x4 d0;
    d0[0] = 1u;                                            // count = 1 (valid descriptor)
    d0[1] = ldsoff;                                        // lds_addr (bytes)
    d0[2] = (unsigned int)(ga & 0xFFFFFFFFull);            // global_addr[31:0]
    d0[3] = (unsigned int)((ga >> 32) & 0x1FFFFFFull)      // global_addr[56:32]
          | 0x80000000u;                                   // type = 2 ("image")

    i32x8 d1;
    d1[0] = 0x00020000;                // workgroup_mask=0, data_size=2 (4 bytes)
    d1[1] = (int)(512u << 16);         // tensor_dim0[15:0] = 512 in bits[31:16]
    d1[2] = (int)(1u   << 16);         // tensor_dim0 hi = 0, tensor_dim1 = 1
    d1[3] = (int)(512u << 16);         // tensor_dim1 hi = 0, tile_dim0 = 512
    d1[4] = 1;                         // tile_dim1 = 1, tile_dim2 = 0
    d1[5] = 512;                       // tensor_dim0_stride[31:0] (elements)
    d1[6] = (int)(512u << 16);         // stride0 hi = 0, tensor_dim1_stride lo = 512
    d1[7] = 0;                         // tensor_dim1_stride hi

    i32x4 z4 = {0, 0, 0, 0};
#if defined(__clang_major__) && (__clang_major__ >= 23)
    i32x8 z8 = {0, 0, 0, 0, 0, 0, 0, 0};
    __builtin_amdgcn_tensor_load_to_lds(d0, d1, z4, z4, z8, 0);
#else
    __builtin_amdgcn_tensor_load_to_lds(d0, d1, z4, z4, 0);
#endif
    __builtin_amdgcn_s_wait_tensorcnt(0);                  // per-wave TENSORcnt drain
  }
  __syncthreads();
#else
  for (int j = threadIdx.x; j < W_; j += 256) g2row[j] = g2[(size_t)row * W_ + j];
  __syncthreads();
#endif

  const float fb = flags[b];
  float acc = 0.f;
  for (int k = 0; k < 2; ++k) {
    const int x = threadIdx.x + k * 256;
    float best = g2row[x];                       // j == x candidate
    for (int d = 1; d < W_; ++d) {               // outward scan, early exit
      const float fd = (float)d;
      const float fd2 = fd * fd;
      if (fd2 >= best) break;                    // no farther j can improve
      const int lo = x - d, hi = x + d;
      if (lo >= 0)  best = fminf(best, fmaf(fd, fd, g2row[lo]));
      if (hi < W_)  best = fminf(best, fmaf(fd, fd, g2row[hi]));
    }
    const float xL = logits[(size_t)row * W_ + x];
    float p = 1.0f / (1.0f + expf(-xL));
    p = fminf(fmaxf(p, 1e-6f), 1.0f - 1e-6f);
    acc += p * sqrtf(best) * fb;
  }

  red[threadIdx.x] = acc;
  __syncthreads();
  for (int s = 128; s > 0; s >>= 1) {
    if (threadIdx.x < s) red[threadIdx.x] += red[threadIdx.x + s];
    __syncthreads();
  }
  if (threadIdx.x == 0) bpart[row] = red[0];
}

// =============================================================================
// Kernel 5: finalize (single wave32). Matrix-core reductions:
// V_WMMA_F32_16X16X4_F32 with all-ones A gives C[m][n] += sum_k B[k][n] in
// full f32 precision. The (k,n) placement of data is irrelevant for a sum, so
// each WMMA consumes 64 floats striped across the wave. All 16 C rows are
// identical, so summing c[0] over all 32 lanes = 2 * total (x0.5 is exact).
// Deterministic: fixed HW accumulation order + fixed xor-butterfly.
// =============================================================================
#if HAVE_WMMA
__device__ __forceinline__ v8f wmma_acc64(const float* __restrict__ src, int base,
                                          int lane, v8f c) {
  v2f a; a.x = 1.0f; a.y = 1.0f;                       // all-ones A (16x4)
  v2f b; b.x = src[base + lane]; b.y = src[base + 32 + lane];  // 64 floats -> B (4x16)
  return __builtin_amdgcn_wmma_f32_16x16x4_f32(false, a, false, b,
                                               (short)0, c, false, false);
}
__device__ __forceinline__ float wmma_extract(v8f c) {
  float v = c[0];                                      // rows 0 and 8 (identical)
  v += __shfl_xor(v, 16, 32);
  v += __shfl_xor(v, 8, 32);
  v += __shfl_xor(v, 4, 32);
  v += __shfl_xor(v, 2, 32);
  v += __shfl_xor(v, 1, 32);
  return v * 0.5f;                                     // exact: two identical rows
}
#endif

__global__ void finalize_kernel(const float* __restrict__ ce_part,
                                const float* __restrict__ inter_part,
                                const float* __restrict__ sump_part,
                                const float* __restrict__ sumt_part,
                                const float* __restrict__ bpart,
                                float* __restrict__ out) {
  const int lane = threadIdx.x;        // blockDim.x == 32: one full wave32
#if HAVE_WMMA
  // boundary: 4096 partials = 64 chained WMMAs
  v8f cb = {};
  for (int ch = 0; ch < 64; ++ch) cb = wmma_acc64(bpart, ch * 64, lane, cb);
  const float b_sum = wmma_extract(cb);

  // ce: 512 partials = 8 chained WMMAs
  v8f cc = {};
  for (int ch = 0; ch < 8; ++ch) cc = wmma_acc64(ce_part, ch * 64, lane, cc);
  const float ce_sum = wmma_extract(cc);

  // dice: each batch's 64 partials fit exactly one WMMA per quantity
  float dsum = 0.f;
  for (int b = 0; b < B_; ++b) {
    v8f ci = {}, cp = {}, ct = {};
    ci = wmma_acc64(inter_part, b * 64, lane, ci);
    cp = wmma_acc64(sump_part,  b * 64, lane, cp);
    ct = wmma_acc64(sumt_part,  b * 64, lane, ct);
    const float inter = wmma_extract(ci);
    const float sp    = wmma_extract(cp);
    const float st    = wmma_extract(ct);
    dsum += (2.0f * inter + 1e-6f) / (sp + st + 1e-6f);
  }

  if (lane == 0) {
    const float ce       = ce_sum / (float)N_;
    const float dice     = 1.0f - dsum / (float)B_;
    const float boundary = b_sum / (float)N_;
    out[0] = ce + dice + 0.1f * boundary;  // total
    out[1] = ce;
    out[2] = dice;
    out[3] = boundary;
  }
#else
  // Scalar fallback (still deterministic; performance irrelevant here).
  if (lane == 0) {
    float ce_sum = 0.f;
    for (int i = 0; i < EW_BLOCKS; ++i) ce_sum += ce_part[i];
    float b_sum = 0.f;
    for (int i = 0; i < ROWS_; ++i) b_sum += bpart[i];
    float dsum = 0.f;
    for (int b = 0; b < B_; ++b) {
      float inter = 0.f, sp = 0.f, st = 0.f;
      for (int i = 0; i < EW_BLOCKS_PER_B; ++i) {
        const int k = b * EW_BLOCKS_PER_B + i;
        inter += inter_part[k];
        sp    += sump_part[k];
        st    += sumt_part[k];
      }
      dsum += (2.0f * inter + 1e-6f) / (sp + st + 1e-6f);
    }
    const float ce       = ce_sum / (float)N_;
    const float dice     = 1.0f - dsum / (float)B_;
    const float boundary = b_sum / (float)N_;
    out[0] = ce + dice + 0.1f * boundary;
    out[1] = ce;
    out[2] = dice;
    out[3] = boundary;
  }
#endif
}

// =============================================================================
// Host launcher
// =============================================================================
extern "C" void kernel_launch(void* const* d_in, const int* in_sizes, int n_in,
                              void* d_out, int out_size, void* d_ws, size_t ws_size,
                              hipStream_t stream) {
  const float* logits = (const float*)d_in[0];
  const int*   tgt    = (const int*)d_in[1];
  float*       out    = (float*)d_out;

  // Workspace layout (floats). Every location is written before being read.
  float* ws         = (float*)d_ws;
  float* g2         = ws;                       // N_ floats (8 MB)
  float* ce_part    = g2 + N_;                  // 512
  float* inter_part = ce_part + EW_BLOCKS;      // 512
  float* sump_part  = inter_part + EW_BLOCKS;   // 512
  float* sumt_part  = sump_part + EW_BLOCKS;    // 512
  float* flags      = sumt_part + EW_BLOCKS;    // 8
  float* bpart      = flags + B_;               // 4096

  edt_cols_kernel<<<(B_ * W_) / 256, 256, 0, stream>>>(tgt, g2);
  ew_reduce_kernel<<<EW_BLOCKS, 256, 0, stream>>>(logits, tgt, ce_part, inter_part,
                                                  sump_part, sumt_part);
  flags_kernel<<<1, 32, 0, stream>>>(sumt_part, flags);
  boundary_kernel<<<ROWS_, 256, 0, stream>>>(logits, g2, flags, bpart);
  finalize_kernel<<<1, 32, 0, stream>>>(ce_part, inter_part, sump_part, sumt_part,
                                        bpart, out);
}